// Conv2dAsLinear_48636209660623
// MI455X (gfx1250) — compile-verified
//
#include <hip/hip_runtime.h>

typedef float v2f __attribute__((ext_vector_type(2)));
typedef float v8f __attribute__((ext_vector_type(8)));

#define KH_   3
#define KW_   3
#define H_    56
#define W_    56
#define CIN   256
#define COUT  256
#define NIMG  32
#define HW_   (H_ * W_)         /* 3136 */
#define MTOT  (NIMG * HW_)      /* 100352 */
#define KTAP  (CIN * KH_ * KW_) /* 2304 */

#define MT  128   /* block tile, spatial  */
#define NT  128   /* block tile, C_out    */
#define KB  16    /* channels staged per round */
#define LDA (KB + 2)  /* 18 floats: conflict-free stride, rows stay 8B-aligned */
#define LDB (KB + 2)
#define NCHUNK ((KH_ * KW_) * (CIN / KB))   /* 9 * 16 = 144 */

__global__ __launch_bounds__(256)
void conv3x3_implicit_gemm_wmma_f32(const float* __restrict__ x,
                                    const float* __restrict__ wgt,
                                    const float* __restrict__ bias,
                                    float* __restrict__ out)
{
    __shared__ float As[2][MT * LDA];   // [buf][spatial][channel]  (M-major, K contiguous)
    __shared__ float Bs[2][NT * LDB];   // [buf][cout][channel]     (N-major, K contiguous)

    const int tid  = threadIdx.x;
    const int lane = tid & 31;
    const int wave = tid >> 5;
    const int wr   = wave & 3;    // wave M offset: 32 rows each
    const int wc   = wave >> 2;   // wave N offset: 64 cols each

    const int mbase = blockIdx.x * MT;
    const int nbase = blockIdx.y * NT;

    // ---- staging coordinates (per thread): 128 rows x 2 channel-halves (8 ch each) ----
    const int s   = tid & (MT - 1);   // row within tile (spatial for A, cout for B)
    const int cg  = tid >> 7;         // 0/1 -> channels [0,8) / [8,16) of the chunk
    const int m   = mbase + s;
    const int img = m / HW_;
    const int rem = m - img * HW_;
    const int oh  = rem / W_;
    const int ow  = rem - oh * W_;
    const float* xin  = x + (size_t)img * CIN * HW_ + (size_t)cg * 8 * HW_;
    const float* wrow = wgt + (size_t)(nbase + s) * KTAP + (size_t)(cg * 8) * (KH_ * KW_);

    // ---- WMMA fragment lane decomposition (16x4 f32 A / 4x16 f32 B layout) ----
    const int lh  = lane & 15;            // row within 16
    const int kh2 = (lane >> 4) << 1;     // K sub-offset 0 or 2 (lane half)

    const v8f vzero = {0.f, 0.f, 0.f, 0.f, 0.f, 0.f, 0.f, 0.f};
    v8f acc[2][4];
    #pragma unroll
    for (int i = 0; i < 2; ++i)
        #pragma unroll
        for (int j = 0; j < 4; ++j)
            acc[i][j] = vzero;

    // ---- stage one 16-channel chunk of A and B into LDS buffer `buf` ----
    auto stage = [&](int chunk, int buf) {
        const int p  = chunk >> 4;          // kernel tap 0..8
        const int c0 = (chunk & 15) << 4;   // channel base 0,16,...,240
        const int kh = p / 3;
        const int kw = p - kh * 3;
        const int ih = oh + kh - 1;
        const int iw = ow + kw - 1;
        const bool valid = ((unsigned)ih < (unsigned)H_) && ((unsigned)iw < (unsigned)W_);
        const float* xs = xin + ih * W_ + iw + (size_t)c0 * HW_;   // + j*HW_ per channel
        const float* ws = wrow + p + (size_t)c0 * (KH_ * KW_);     // + j*9  per channel

        float av[8], bv[8];
        #pragma unroll
        for (int j = 0; j < 8; ++j)
            av[j] = valid ? xs[(size_t)j * HW_] : 0.f;
        #pragma unroll
        for (int j = 0; j < 8; ++j)
            bv[j] = ws[(size_t)j * (KH_ * KW_)];
        #pragma unroll
        for (int j = 0; j < 8; ++j)
            As[buf][s * LDA + cg * 8 + j] = av[j];
        #pragma unroll
        for (int j = 0; j < 8; ++j)
            Bs[buf][s * LDB + cg * 8 + j] = bv[j];
    };

    // ---- 4 k-steps of 4, 8 WMMAs each, from LDS buffer `buf` ----
    auto compute = [&](int buf) {
        #pragma unroll
        for (int k0 = 0; k0 < KB; k0 += 4) {
            const int kc = k0 + kh2;
            v2f a0 = *(const v2f*)&As[buf][(wr * 32 +  0 + lh) * LDA + kc];
            v2f a1 = *(const v2f*)&As[buf][(wr * 32 + 16 + lh) * LDA + kc];
            v2f b0 = *(const v2f*)&Bs[buf][(wc * 64 +  0 + lh) * LDB + kc];
            v2f b1 = *(const v2f*)&Bs[buf][(wc * 64 + 16 + lh) * LDB + kc];
            v2f b2 = *(const v2f*)&Bs[buf][(wc * 64 + 32 + lh) * LDB + kc];
            v2f b3 = *(const v2f*)&Bs[buf][(wc * 64 + 48 + lh) * LDB + kc];

            acc[0][0] = __builtin_amdgcn_wmma_f32_16x16x4_f32(false, a0, false, b0, (short)0, acc[0][0], false, false);
            acc[0][1] = __builtin_amdgcn_wmma_f32_16x16x4_f32(false, a0, false, b1, (short)0, acc[0][1], false, false);
            acc[0][2] = __builtin_amdgcn_wmma_f32_16x16x4_f32(false, a0, false, b2, (short)0, acc[0][2], false, false);
            acc[0][3] = __builtin_amdgcn_wmma_f32_16x16x4_f32(false, a0, false, b3, (short)0, acc[0][3], false, false);
            acc[1][0] = __builtin_amdgcn_wmma_f32_16x16x4_f32(false, a1, false, b0, (short)0, acc[1][0], false, false);
            acc[1][1] = __builtin_amdgcn_wmma_f32_16x16x4_f32(false, a1, false, b1, (short)0, acc[1][1], false, false);
            acc[1][2] = __builtin_amdgcn_wmma_f32_16x16x4_f32(false, a1, false, b2, (short)0, acc[1][2], false, false);
            acc[1][3] = __builtin_amdgcn_wmma_f32_16x16x4_f32(false, a1, false, b3, (short)0, acc[1][3], false, false);
        }
    };

    // ---- software-pipelined main loop: one barrier per chunk ----
    stage(0, 0);
    __syncthreads();
    #pragma unroll 2
    for (int ch = 0; ch < NCHUNK; ++ch) {
        const int cur = ch & 1;
        if (ch + 1 < NCHUNK)
            stage(ch + 1, cur ^ 1);   // fill next buffer while computing current
        compute(cur);
        __syncthreads();
    }

    // ---- epilogue: bias add + scatter to NCHW ----
    // C/D 16x16 f32 layout: VGPR r, lanes 0-15 -> M=r, lanes 16-31 -> M=r+8; N = lane%16
    const int mrow_off = (lane >> 4) << 3;   // +8 for upper lane half
    #pragma unroll
    for (int ni = 0; ni < 4; ++ni) {
        const int o  = nbase + wc * 64 + ni * 16 + lh;
        const float bv = bias[o];
        #pragma unroll
        for (int mi = 0; mi < 2; ++mi) {
            #pragma unroll
            for (int r = 0; r < 8; ++r) {
                const int mm = mbase + wr * 32 + mi * 16 + mrow_off + r;
                const int im = mm / HW_;
                const int rm = mm - im * HW_;
                out[((size_t)im * COUT + o) * HW_ + rm] = acc[mi][ni][r] + bv;
            }
        }
    }
}

extern "C" void kernel_launch(void* const* d_in, const int* in_sizes, int n_in,
                              void* d_out, int out_size, void* d_ws, size_t ws_size,
                              hipStream_t stream) {
    (void)in_sizes; (void)n_in; (void)out_size; (void)d_ws; (void)ws_size;
    const float* x    = (const float*)d_in[0];
    const float* wgt  = (const float*)d_in[1];
    const float* bias = (const float*)d_in[2];
    float* out        = (float*)d_out;

    dim3 grid(MTOT / MT, COUT / NT);   // 784 x 2
    dim3 block(256);
    hipLaunchKernelGGL(conv3x3_implicit_gemm_wmma_f32, grid, block, 0, stream,
                       x, wgt, bias, out);
}